// FeaturePropogationModule_7730941133288
// MI455X (gfx1250) — compile-verified
//
#include <hip/hip_runtime.h>

typedef float v2f __attribute__((ext_vector_type(2)));
typedef float v8f __attribute__((ext_vector_type(8)));

__device__ __forceinline__ v8f wmma_f32(v2f a, v2f b, v8f c) {
  // D = A(16x4 f32) x B(4x16 f32) + C(16x16 f32)
  return __builtin_amdgcn_wmma_f32_16x16x4_f32(false, a, false, b, (short)0, c, false, false);
}

__device__ __forceinline__ void st2(float* d, float a, float b) {
  v2f t; t.x = a; t.y = b; *(v2f*)d = t;
}

// ---- LDS layout (float offsets) ----
// Hot, small buffers low so DS immediates fold; W matrices pair-interleaved:
//   element (k, n) lives at  (k>>1)*STRIDE + n*2 + (k&1)
// so a B fragment (W[k][n], W[k+1][n]) is one aligned ds_load_b64.
// Pair-row strides (288, 160) shift lanes 16-31 (pair p+1) by 32 banks -> conflict-free.
#define OFF_A    0        // A_hat: 16 rows x stride 20
#define OFF_B1   320      // b1: 128
#define OFF_B2   448      // b2: 64
#define OFF_X    512      // X: 2 elems x 16 rows x stride 260 (row-major, A-frags)
#define OFF_H    8832     // H: 2 x 16 x stride 132            (row-major, A-frags)
#define OFF_ZP   13056    // Z: 2 x (8 pairs x stride 288)     (pair-interleaved, B-frags)
#define OFF_YP   17664    // Y: 2 x (8 pairs x stride 160)     (pair-interleaved, B-frags)
#define OFF_W2   20224    // W2: 64 pairs x stride 160
#define OFF_W1   30464    // W1: 128 pairs x stride 288
#define LDS_FLOATS 67328  // 269,312 bytes < 320 KB

__global__ __launch_bounds__(256) void gcn2_fused(
    const float* __restrict__ fea,
    const float* __restrict__ gW1,
    const float* __restrict__ gb1,
    const float* __restrict__ gW2,
    const float* __restrict__ gb2,
    float* __restrict__ out,
    int Btot, int batchPerWg)
{
  extern __shared__ float smem[];
  float* sA   = smem + OFF_A;
  float* sB1  = smem + OFF_B1;
  float* sB2  = smem + OFF_B2;
  float* sX   = smem + OFF_X;
  float* sH   = smem + OFF_H;
  float* sZp  = smem + OFF_ZP;
  float* sYp  = smem + OFF_YP;
  float* sW2p = smem + OFF_W2;
  float* sW1p = smem + OFF_W1;

  const int tid  = threadIdx.x;
  const int lane = tid & 31;
  const int wave = tid >> 5;
  const int e    = wave >> 2;     // batch element of the pair (0/1)
  const int sub  = wave & 3;      // sub-wave id within element
  const int m    = lane & 15;     // M row (A-frag) / N col (B,C,D frags)
  const int hi   = lane >> 4;     // lane-half
  const int koff = hi << 1;       // A-frag K offset: lanes 0-15 K{0,1}, 16-31 K{2,3}

  // ---------- one-time staging ----------
  for (int i = tid; i < 320;  i += 256) sA[i] = 0.0f;
  for (int i = tid; i < 8320; i += 256) sX[i] = 0.0f;     // incl. pad rows 14,15

  // W1 -> pair-interleaved LDS (128 pairs x 32 col4 chunks)
  for (int i = tid; i < 4096; i += 256) {
    int p = i >> 5, c = (i & 31) << 2;
    float4 g0 = *(const float4*)(gW1 + (size_t)(2*p + 0)*128 + c);
    float4 g1 = *(const float4*)(gW1 + (size_t)(2*p + 1)*128 + c);
    float* d = sW1p + p*288 + c*2;
    st2(d + 0, g0.x, g1.x); st2(d + 2, g0.y, g1.y);
    st2(d + 4, g0.z, g1.z); st2(d + 6, g0.w, g1.w);
  }
  // W2 -> pair-interleaved LDS (64 pairs x 16 col4 chunks)
  for (int i = tid; i < 1024; i += 256) {
    int p = i >> 4, c = (i & 15) << 2;
    float4 g0 = *(const float4*)(gW2 + (size_t)(2*p + 0)*64 + c);
    float4 g1 = *(const float4*)(gW2 + (size_t)(2*p + 1)*64 + c);
    float* d = sW2p + p*160 + c*2;
    st2(d + 0, g0.x, g1.x); st2(d + 2, g0.y, g1.y);
    st2(d + 4, g0.z, g1.z); st2(d + 6, g0.w, g1.w);
  }
  if (tid < 128) sB1[tid] = gb1[tid];
  if (tid < 64)  sB2[tid] = gb2[tid];
  __syncthreads();

  if (tid == 0) {
    // A_hat = D^{-1/2} (A + I) D^{-1/2} on the fixed 14-node arch graph.
    const int ord[14] = {6,5,4,3,2,1,0,7,8,9,10,11,12,13};
    for (int k = 0; k < 13; ++k) {
      int s = ord[k], t = ord[k+1];
      sA[s*20 + t] = 1.f; sA[t*20 + s] = 1.f;
    }
    for (int i = 0; i < 7; ++i) { sA[i*20 + (i+7)] = 1.f; sA[(i+7)*20 + i] = 1.f; }
    for (int i = 0; i < 14; ++i) sA[i*20 + i] = 1.f;
    for (int i = 0; i < 14; ++i) {
      float di = (i==0||i==6||i==7||i==13) ? 0.57735026918962576f : 0.5f;   // deg 3 / 4
      for (int j = 0; j < 14; ++j) {
        float dj = (j==0||j==6||j==7||j==13) ? 0.57735026918962576f : 0.5f;
        sA[i*20 + j] *= di * dj;
      }
    }
  }
  __syncthreads();

  // ---- iteration-invariant tile bases (pure DS-immediate addressing inside loops) ----
  const int nb0 = sub*32, nb1 = nb0 + 16, nb2 = sub*16;
  float* He = sH + e*2112;
  const float* Xr  = sX   + e*4160 + m*260;
  const float* Hr  = He   + m*132;
  const float* Ar  = sA   + m*20;
  const float* Wp0 = sW1p + hi*288 + (nb0 + m)*2;
  const float* Wp1 = sW1p + hi*288 + (nb1 + m)*2;
  const float* Vp  = sW2p + hi*160 + (nb2 + m)*2;
  const float* Zb0 = sZp  + e*2304 + hi*288 + (nb0 + m)*2;
  const float* Zb1 = sZp  + e*2304 + hi*288 + (nb1 + m)*2;
  const float* Yb  = sYp  + e*1280 + hi*160 + (nb2 + m)*2;
  float* Zst0 = sZp + e*2304 + (4*hi)*288 + (nb0 + m)*2;
  float* Zst1 = sZp + e*2304 + (4*hi)*288 + (nb1 + m)*2;
  float* Yst  = sYp + e*1280 + (4*hi)*160 + (nb2 + m)*2;

  for (int it = 0; it < batchPerWg; it += 2) {
    const int b0 = blockIdx.x * batchPerWg + it;

    // ---- stage X pair via async global->LDS (no VGPR data movement) ----
    for (int i = tid; i < 2048; i += 256) {
      int ee = i >> 10, rr = (i >> 6) & 15, cc = (i & 63) << 2;
      int b = b0 + ee;
      if (rr < 14 && b < Btot) {
        const float* gp = fea + ((size_t)b*14 + rr)*256 + cc;
        unsigned loff = (unsigned)(uintptr_t)(sX + ee*4160 + rr*260 + cc);
        asm volatile("global_load_async_to_lds_b128 %0, %1, off"
                     :: "v"(loff), "v"(gp) : "memory");
        if (b + 2 < Btot) __builtin_prefetch(gp + (size_t)2*14*256, 0, 1);
      }
    }
    asm volatile("s_wait_asynccnt 0x0" ::: "memory");
    __syncthreads();

    // ---- GEMM1: Z[16,128] = X[16,256] @ W1, 2 n-tiles / wave ----
    v8f z0 = {}; v8f z1 = {};
    #pragma unroll
    for (int kc = 0; kc < 4; ++kc) {              // chunked: DS offsets stay immediate
      const float* Xc  = Xr  + kc*64;
      const float* Wc0 = Wp0 + kc*32*288;
      const float* Wc1 = Wp1 + kc*32*288;
      #pragma unroll
      for (int kk = 0; kk < 64; kk += 4) {
        v2f a   = *(const v2f*)(Xc  + kk + koff);
        v2f bb0 = *(const v2f*)(Wc0 + (kk >> 1)*288);
        v2f bb1 = *(const v2f*)(Wc1 + (kk >> 1)*288);
        z0 = wmma_f32(a, bb0, z0);
        z1 = wmma_f32(a, bb1, z1);
      }
    }
    #pragma unroll
    for (int t = 0; t < 4; ++t) {                 // rows 2t,2t+1 -> one b64 pair slot
      st2(Zst0 + t*288, z0[2*t], z0[2*t + 1]);
      st2(Zst1 + t*288, z1[2*t], z1[2*t + 1]);
    }
    __syncthreads();

    // ---- Mix1: H = relu(A_hat @ Z + b1) ----
    v8f h0 = {}; v8f h1 = {};
    #pragma unroll
    for (int k0 = 0; k0 < 16; k0 += 4) {
      v2f a   = *(const v2f*)(Ar  + k0 + koff);
      v2f bb0 = *(const v2f*)(Zb0 + (k0 >> 1)*288);
      v2f bb1 = *(const v2f*)(Zb1 + (k0 >> 1)*288);
      h0 = wmma_f32(a, bb0, h0);
      h1 = wmma_f32(a, bb1, h1);
    }
    {
      float c0 = sB1[nb0 + m], c1 = sB1[nb1 + m];
      #pragma unroll
      for (int j = 0; j < 8; ++j) {               // D layout: row = j + 8*hi, col = m
        int r = j + 8*hi;
        He[r*132 + nb0 + m] = fmaxf(h0[j] + c0, 0.f);
        He[r*132 + nb1 + m] = fmaxf(h1[j] + c1, 0.f);
      }
    }
    __syncthreads();

    // ---- GEMM2: Y[16,64] = H[16,128] @ W2, 1 n-tile / wave ----
    v8f y = {};
    #pragma unroll
    for (int k0 = 0; k0 < 128; k0 += 4) {
      v2f a  = *(const v2f*)(Hr + k0 + koff);
      v2f bb = *(const v2f*)(Vp + (k0 >> 1)*160);
      y = wmma_f32(a, bb, y);
    }
    #pragma unroll
    for (int t = 0; t < 4; ++t)
      st2(Yst + t*160, y[2*t], y[2*t + 1]);
    __syncthreads();

    // ---- Mix2: out = A_hat @ Y + b2, masked store of 14 real rows ----
    v8f o = {};
    #pragma unroll
    for (int k0 = 0; k0 < 16; k0 += 4) {
      v2f a  = *(const v2f*)(Ar + k0 + koff);
      v2f bb = *(const v2f*)(Yb + (k0 >> 1)*160);
      o = wmma_f32(a, bb, o);
    }
    {
      float c2 = sB2[nb2 + m];
      if (b0 + e < Btot) {
        float* op = out + ((size_t)(b0 + e)*14)*64 + nb2 + m;
        #pragma unroll
        for (int j = 0; j < 8; ++j) {
          int r = j + 8*hi;
          if (r < 14) op[r*64] = o[j] + c2;
        }
      }
    }
    __syncthreads();
  }
}

extern "C" void kernel_launch(void* const* d_in, const int* in_sizes, int n_in,
                              void* d_out, int out_size, void* d_ws, size_t ws_size,
                              hipStream_t stream) {
  const float* fea = (const float*)d_in[0];
  const float* W1  = (const float*)d_in[1];
  const float* b1  = (const float*)d_in[2];
  const float* W2  = (const float*)d_in[3];
  const float* b2  = (const float*)d_in[4];
  float* out = (float*)d_out;

  const int B = in_sizes[0] / (14 * 256);     // 16384
  const int BATCH_PER_WG = 16;                // W1/W2 staged once per WG
  const int grid = (B + BATCH_PER_WG - 1) / BATCH_PER_WG;
  const size_t shmem = (size_t)LDS_FLOATS * sizeof(float);

  gcn2_fused<<<dim3(grid), dim3(256), shmem, stream>>>(fea, W1, b1, W2, b2, out,
                                                       B, BATCH_PER_WG);
}